// EdgeConv2d_69509750718743
// MI455X (gfx1250) — compile-verified
//
#include <hip/hip_runtime.h>

typedef float v8f __attribute__((ext_vector_type(8)));
typedef float v2f __attribute__((ext_vector_type(2)));

#define BB 4
#define CC 128
#define NN 4096
#define KK 16
#define CO 128

// ---------------------------------------------------------------------------
// Stage 1: per-node transform  Y[b, n, o] = sum_c W[o,c] * x[b, base(o)+c, n]
// base(o) = ((o>>5)&1)*64  (groups 0,1 read x[0:64]/x[64:128]; groups 2,3 same)
// One wave computes a 16(n) x 16(o) tile of Y via 16x v_wmma_f32_16x16x4_f32.
// A fragment = x tile (M=n, K=c), B fragment = weight tile (K=c, N=o).
// Y layout is node-major [B, N, 128] so stage 2 gathers contiguous columns.
// ---------------------------------------------------------------------------
__global__ __launch_bounds__(256) void edgeconv_stage1_wmma(
    const float* __restrict__ x, const float* __restrict__ w,
    float* __restrict__ Y) {
  const int lane = threadIdx.x & 31;
  const int wave = threadIdx.x >> 5;
  const int b    = blockIdx.z;
  const int o0   = blockIdx.y << 4;                 // 8 o-tiles of 16
  const int n0   = (blockIdx.x * 8 + wave) << 4;    // 8 waves * 16 n each
  const int base = ((o0 >> 5) & 1) << 6;            // input channel base
  const int m    = lane & 15;                       // A: M index / B: N index
  const int koff = (lane >> 4) << 1;                // lanes 16-31 hold K+2,K+3

  const float* xb = x + (size_t)b * CC * NN;
  v8f acc = {0.f, 0.f, 0.f, 0.f, 0.f, 0.f, 0.f, 0.f};

#pragma unroll
  for (int kk = 0; kk < 16; ++kk) {                 // 64 input ch / K=4
    const int c0 = kk * 4 + koff;
    v2f a, bm;
    // A (16x4 f32): element (M=n0+m, K=c0[+1])
    a.x = xb[(size_t)(base + c0)     * NN + n0 + m];
    a.y = xb[(size_t)(base + c0 + 1) * NN + n0 + m];
    // B (4x16 f32): element (K=c0[+1], N=o0+m)
    bm.x = w[(o0 + m) * 64 + c0];
    bm.y = w[(o0 + m) * 64 + c0 + 1];
    acc = __builtin_amdgcn_wmma_f32_16x16x4_f32(
        /*neg_a=*/false, a, /*neg_b=*/false, bm,
        /*c_mod=*/(short)0, acc, /*reuse_a=*/false, /*reuse_b=*/false);
  }

  // D (16x16 f32): lane l, vgpr v -> row M = v + 8*(l>>4), col N = l&15
  float* yb = Y + (size_t)b * NN * CO;
  const int nhi = (lane >> 4) << 3;
#pragma unroll
  for (int v = 0; v < 8; ++v) {
    yb[(size_t)(n0 + nhi + v) * CO + (o0 + m)] = acc[v];  // 64B runs per 16 lanes
  }
}

// ---------------------------------------------------------------------------
// Stage 2: gather + relu + mean over K.
//   o <  64: out[b,o,n] = mean_k relu( Y[b, i_k, o] + bias[o] )
//   o >= 64: out[b,o,n] = mean_k relu( Y[b, j_k, o] - Y[b, i_k, o] + bias[o] )
// thread = (quarter q of 32 channels, node n). Lanes vary n -> coalesced
// stores; gathered Y columns are contiguous 128B float4 runs per lane.
// q is wave-uniform (q = tid>>6), so the i-only / i+j branch never diverges.
// ---------------------------------------------------------------------------
__global__ __launch_bounds__(256) void edgeconv_stage2_gather(
    const float* __restrict__ Y, const long long* __restrict__ edge,
    const float* __restrict__ bias, float* __restrict__ out) {
  const int q  = threadIdx.x >> 6;          // 0..3 -> 32-channel quarter
  const int nl = threadIdx.x & 63;
  const int b  = blockIdx.y;
  const int n  = blockIdx.x * 64 + nl;
  const int ob = q * 32;

  const long long* e0 = edge + ((size_t)b * NN + n) * KK;   // edge_index[0] = j
  const long long* e1 = e0 + (size_t)BB * NN * KK;          // edge_index[1] = i
  const float* yb = Y + (size_t)b * NN * CO;

  float4 bv[8];
#pragma unroll
  for (int c = 0; c < 8; ++c) bv[c] = ((const float4*)(bias + ob))[c];
  float4 acc[8];
#pragma unroll
  for (int c = 0; c < 8; ++c) acc[c] = make_float4(0.f, 0.f, 0.f, 0.f);

  for (int k = 0; k < KK; ++k) {
    const int i = (int)e1[k];
    const float4* yi = (const float4*)(yb + (size_t)i * CO + ob);
    if (q < 2) {                             // channels from x_i only
#pragma unroll
      for (int c = 0; c < 8; ++c) {
        float4 t = yi[c];
        acc[c].x += fmaxf(t.x + bv[c].x, 0.f);
        acc[c].y += fmaxf(t.y + bv[c].y, 0.f);
        acc[c].z += fmaxf(t.z + bv[c].z, 0.f);
        acc[c].w += fmaxf(t.w + bv[c].w, 0.f);
      }
    } else {                                 // channels from (x_j - x_i)
      const int j = (int)e0[k];
      const float4* yj = (const float4*)(yb + (size_t)j * CO + ob);
#pragma unroll
      for (int c = 0; c < 8; ++c) {
        float4 ti = yi[c], tj = yj[c];
        acc[c].x += fmaxf(tj.x - ti.x + bv[c].x, 0.f);
        acc[c].y += fmaxf(tj.y - ti.y + bv[c].y, 0.f);
        acc[c].z += fmaxf(tj.z - ti.z + bv[c].z, 0.f);
        acc[c].w += fmaxf(tj.w - ti.w + bv[c].w, 0.f);
      }
    }
  }

  const float s = 1.0f / (float)KK;
  float* op = out + ((size_t)b * CO + ob) * NN + n;  // out[b, o, n, 1]
#pragma unroll
  for (int c = 0; c < 8; ++c) {
    op[(size_t)(4 * c + 0) * NN] = acc[c].x * s;
    op[(size_t)(4 * c + 1) * NN] = acc[c].y * s;
    op[(size_t)(4 * c + 2) * NN] = acc[c].z * s;
    op[(size_t)(4 * c + 3) * NN] = acc[c].w * s;
  }
}

// ---------------------------------------------------------------------------
// Fallback (workspace too small): direct per-(b,o,n) evaluation, no Y table.
// ---------------------------------------------------------------------------
__global__ __launch_bounds__(256) void edgeconv_fallback(
    const float* __restrict__ x, const long long* __restrict__ edge,
    const float* __restrict__ w, const float* __restrict__ bias,
    float* __restrict__ out) {
  const int idx = blockIdx.x * 256 + threadIdx.x;
  if (idx >= BB * CO * NN) return;
  const int n = idx & (NN - 1);
  const int o = (idx >> 12) & (CO - 1);
  const int b = idx >> 19;
  const int base = ((o >> 5) & 1) << 6;

  const long long* e0 = edge + ((size_t)b * NN + n) * KK;
  const long long* e1 = e0 + (size_t)BB * NN * KK;
  const float* xb = x + (size_t)b * CC * NN;
  const float* wr = w + o * 64;

  float acc = 0.f;
  for (int k = 0; k < KK; ++k) {
    const int i = (int)e1[k];
    float t = 0.f;
    if (o < 64) {
      for (int c = 0; c < 64; ++c)
        t += wr[c] * xb[(size_t)(base + c) * NN + i];
    } else {
      const int j = (int)e0[k];
      for (int c = 0; c < 64; ++c)
        t += wr[c] * (xb[(size_t)(base + c) * NN + j] -
                      xb[(size_t)(base + c) * NN + i]);
    }
    acc += fmaxf(t + bias[o], 0.f);
  }
  out[idx] = acc * (1.0f / (float)KK);
}

extern "C" void kernel_launch(void* const* d_in, const int* in_sizes, int n_in,
                              void* d_out, int out_size, void* d_ws, size_t ws_size,
                              hipStream_t stream) {
  const float*     x    = (const float*)d_in[0];      // [4,128,4096,1] f32
  const long long* edge = (const long long*)d_in[1];  // [2,4,4096,16] i64
  const float*     w    = (const float*)d_in[2];      // [128,64,1,1] f32
  const float*     bias = (const float*)d_in[3];      // [128] f32
  float*           out  = (float*)d_out;              // [4,128,4096,1] f32

  const size_t needY = (size_t)BB * NN * CO * sizeof(float);  // 8 MB
  if (ws_size >= needY) {
    float* Y = (float*)d_ws;  // node-major [B, N, 128]
    dim3 g1(NN / (8 * 16), CO / 16, BB);  // (32, 8, 4), 8 waves/block
    edgeconv_stage1_wmma<<<g1, 256, 0, stream>>>(x, w, Y);
    dim3 g2(NN / 64, BB);                 // (64, 4)
    edgeconv_stage2_gather<<<g2, 256, 0, stream>>>(Y, edge, bias, out);
  } else {
    const int total = BB * CO * NN;
    edgeconv_fallback<<<(total + 255) / 256, 256, 0, stream>>>(x, edge, w, bias, out);
  }
}